// vapl_grid_base_14070312861787
// MI455X (gfx1250) — compile-verified
//
#include <hip/hip_runtime.h>
#include <hip/hip_bf16.h>
#include <stdint.h>

// HashGrid VPL sampling, MI455X (gfx1250, wave32).
// Optimization: 27-stencil x 8-corner = 216 lookups collapses to the 64 unique
// lattice points {-1,0,1,2}^3 with separable weights {1-w, 1, 1, w} per axis
// (exact because block = 1/RES is a power of two). Tables (64MB + 128MB) fit
// the 192MB L2, so gathers are L2-resident; a prefetch pass (global_prefetch_b8,
// no LOADcnt, no VGPR dest) runs ahead of the b128 gather pass.

#define HP2 2654435761u
#define HP3 805459861u
#define TBL_MASK 0x3FFFFFu   // TABLE_SIZE = 1<<22

__device__ __forceinline__ float fsigmoid(float x) {
    return 1.0f / (1.0f + __expf(-x));
}

__global__ __launch_bounds__(256) void vapl_grid_kernel(
    const float* __restrict__ pos,
    const float* __restrict__ gtab,   // [2^22, 4]
    const float* __restrict__ vtab,   // [2^22, 8]
    float* __restrict__ gout,         // [N, 4]
    float* __restrict__ vout,         // [N, 7]
    int npts)
{
    const int n = blockIdx.x * blockDim.x + threadIdx.x;
    if (n >= npts) return;

    // npos = (pos - (-10)) / 20 ; x = npos * RES (RES = 128)
    const float px = (pos[3 * n + 0] + 10.0f) * 0.05f * 128.0f;
    const float py = (pos[3 * n + 1] + 10.0f) * 0.05f * 128.0f;
    const float pz = (pos[3 * n + 2] + 10.0f) * 0.05f * 128.0f;

    const float fx = floorf(px), fy = floorf(py), fz = floorf(pz);
    const float wx = px - fx,   wy = py - fy,   wz = pz - fz;
    const int   bx = (int)fx,   by = (int)fy,   bz = (int)fz;

    // Separable hash terms for lattice offsets {-1,0,1,2} (uint32 wraparound
    // matches the reference's int32 -> uint32 cast for negative coords).
    uint32_t hx[4], hy[4], hz[4];
#pragma unroll
    for (int d = 0; d < 4; ++d) {
        hx[d] = (uint32_t)(bx + d - 1);
        hy[d] = (uint32_t)(by + d - 1) * HP2;
        hz[d] = (uint32_t)(bz + d - 1) * HP3;
    }
    // Collapsed stencil weights per axis: offset -1 -> 1-w, 0 -> 1, 1 -> 1, 2 -> w
    const float Wx[4] = {1.0f - wx, 1.0f, 1.0f, wx};
    const float Wy[4] = {1.0f - wy, 1.0f, 1.0f, wy};
    const float Wz[4] = {1.0f - wz, 1.0f, 1.0f, wz};

    // ---- Prefetch pass: 64 combos x 2 tables. global_prefetch_b8 carries no
    // VGPR destination and does not touch LOADcnt, so this puts far more line
    // fetches in flight than the 63-deep load counter would allow.
#pragma unroll
    for (int i = 0; i < 4; ++i) {
#pragma unroll
        for (int j = 0; j < 4; ++j) {
            const uint32_t hxy = hx[i] ^ hy[j];
#pragma unroll
            for (int k = 0; k < 4; ++k) {
                const uint32_t idx = (hxy ^ hz[k]) & TBL_MASK;
                __builtin_prefetch(gtab + ((size_t)idx << 2), 0, 3);
                __builtin_prefetch(vtab + ((size_t)idx << 3), 0, 3);
            }
        }
    }

    // ---- Gather + accumulate pass (global_load_b128, rows are 16B/32B aligned)
    float g0 = 0.f, g1 = 0.f, g2 = 0.f, g3 = 0.f;
    float v0 = 0.f, v1 = 0.f, v2 = 0.f, v3 = 0.f;
    float v4 = 0.f, v5 = 0.f, v6 = 0.f, v7 = 0.f;

#pragma unroll
    for (int i = 0; i < 4; ++i) {
#pragma unroll
        for (int j = 0; j < 4; ++j) {
            const float    wxy = Wx[i] * Wy[j];
            const uint32_t hxy = hx[i] ^ hy[j];
#pragma unroll
            for (int k = 0; k < 4; ++k) {
                const float    w   = wxy * Wz[k];
                const uint32_t idx = (hxy ^ hz[k]) & TBL_MASK;

                const float4 g4 = *(const float4*)(gtab + ((size_t)idx << 2));
                const float4* vp = (const float4*)(vtab + ((size_t)idx << 3));
                const float4 va = vp[0];
                const float4 vb = vp[1];

                g0 = fmaf(w, g4.x, g0);
                g1 = fmaf(w, g4.y, g1);
                g2 = fmaf(w, g4.z, g2);
                g3 = fmaf(w, g4.w, g3);
                v0 = fmaf(w, va.x, v0);
                v1 = fmaf(w, va.y, v1);
                v2 = fmaf(w, va.z, v2);
                v3 = fmaf(w, va.w, v3);
                v4 = fmaf(w, vb.x, v4);
                v5 = fmaf(w, vb.y, v5);
                v6 = fmaf(w, vb.z, v6);
                v7 = fmaf(w, vb.w, v7);
            }
        }
    }

    // ---- encode ----
    // mean = sigmoid(g[0:3]) * (BB_MAX-BB_MIN) + BB_MIN = sigmoid * 20 - 10
    const float m0 = fsigmoid(g0) * 20.0f - 10.0f;
    const float m1 = fsigmoid(g1) * 20.0f - 10.0f;
    const float m2 = fsigmoid(g2) * 20.0f - 10.0f;
    // softplus(g3), numerically stable
    const float var = fmaxf(g3, 0.0f) + log1pf(__expf(-fabsf(g3)));

    const float sharp = __expf(v0);
    const float nrm   = sqrtf(v1 * v1 + v2 * v2 + v3 * v3);
    const float inv   = 1.0f / fmaxf(nrm, 1e-12f);
    const float a0 = fsigmoid(v4);
    const float a1 = fsigmoid(v5);
    const float a2 = fsigmoid(v6);

    float4 go;
    go.x = m0; go.y = m1; go.z = m2; go.w = var;
    *(float4*)(gout + ((size_t)n << 2)) = go;   // 16B-aligned vector store

    float* vo = vout + (size_t)n * 7;
    vo[0] = sharp;
    vo[1] = v1 * inv;
    vo[2] = v2 * inv;
    vo[3] = v3 * inv;
    vo[4] = a0;
    vo[5] = a1;
    vo[6] = a2;
}

extern "C" void kernel_launch(void* const* d_in, const int* in_sizes, int n_in,
                              void* d_out, int out_size, void* d_ws, size_t ws_size,
                              hipStream_t stream) {
    const float* pos  = (const float*)d_in[0];   // [N, 3]
    const float* gtab = (const float*)d_in[1];   // [2^22, 4]
    const float* vtab = (const float*)d_in[2];   // [2^22, 8]

    const int npts = in_sizes[0] / 3;
    float* gout = (float*)d_out;                       // [N,4] first
    float* vout = (float*)d_out + (size_t)npts * 4;    // then [N,7]

    const int block = 256;                             // 8 wave32 waves
    const int grid  = (npts + block - 1) / block;
    vapl_grid_kernel<<<grid, block, 0, stream>>>(pos, gtab, vtab, gout, vout, npts);
}